// SlidingWindowLSTM_34187939676196
// MI455X (gfx1250) — compile-verified
//
#include <hip/hip_runtime.h>

// Fused 2-layer LSTM (H1=16, H2=10, IN=32, T=512, B=4096) + FC head for gfx1250.
// One wave (32 lanes) <-> one tile of 16 batch rows; 256 workgroups total.
// All matmuls use V_WMMA_F32_16X16X4_F32; C<->A layout transpose via LDS.
// Sigmoid gates use pre-scaled weights (z/2) so sigmoid = fma(0.5, tanh, 0.5).

typedef float v2f __attribute__((ext_vector_type(2)));
typedef float v8f __attribute__((ext_vector_type(8)));

#define B_TOT 4096
#define T_LEN 512
#define IN_SZ 32
#define H1_SZ 16
#define H2_SZ 10

#if __has_builtin(__builtin_amdgcn_tanhf)
#define HAVE_NATIVE_TANH 1
#else
#define HAVE_NATIVE_TANH 0
#endif

__device__ __forceinline__ float fast_tanh(float x) {
#if HAVE_NATIVE_TANH
    return __builtin_amdgcn_tanhf(x);        // v_tanh_f32
#else
    float e = __expf(2.0f * x);
    return 1.0f - 2.0f * __builtin_amdgcn_rcpf(e + 1.0f);
#endif
}

// Input zs is ALREADY z/2 (weights/bias pre-scaled by 0.5): sigmoid(z)=0.5*tanh(z/2)+0.5
__device__ __forceinline__ float sigmoid_from_half(float zs) {
#if HAVE_NATIVE_TANH
    return __builtin_fmaf(0.5f, __builtin_amdgcn_tanhf(zs), 0.5f);
#else
    return __builtin_amdgcn_rcpf(1.0f + __expf(-2.0f * zs));
#endif
}

__global__ __launch_bounds__(32) void lstm2_wmma_kernel(
    const float* __restrict__ x,
    const float* __restrict__ w_ih1, const float* __restrict__ w_hh1,
    const float* __restrict__ b_ih1, const float* __restrict__ b_hh1,
    const float* __restrict__ w_ih2, const float* __restrict__ w_hh2,
    const float* __restrict__ b_ih2, const float* __restrict__ b_hh2,
    const float* __restrict__ fc1_w, const float* __restrict__ fc1_b,
    const float* __restrict__ fc2_w, const float* __restrict__ fc2_b,
    float* __restrict__ out)
{
    __shared__ float lds1[16][20];   // h1 tile, row = batch m, col = hidden n (pad 20 -> conflict free)
    __shared__ float lds2[16][20];   // h2 tile (cols 10..15 kept zero)

    const int lane = threadIdx.x;
    const int col  = lane & 15;      // N (or M for A-frags)
    const int kh   = lane >> 4;      // lane-half selects K+2/K+3 in fragments

    // Zero LDS (single wave; DS ops are in-order within a wave, no barrier needed)
    for (int i = lane; i < 16 * 20; i += 32) {
        (&lds1[0][0])[i] = 0.0f;
        (&lds2[0][0])[i] = 0.0f;
    }

    // ---- Preload weight fragments (B-matrix layout: v0 = W^T[k][n], v1 = W^T[k+1][n],
    //      k = 4*kc + 2*kh, n = tile*16 + col). B[k][n] = W[n][k] since gates = h @ W^T.
    //      Gate tiles nt=0(i),1(f),3(o) pre-scaled by 0.5 for the tanh-based sigmoid.
    v2f Bx1[8][4];   // W_ih1: K=32 (8 chunks), N=64 (4 gate tiles of 16)
    v2f Bh1[4][4];   // W_hh1: K=16
    v2f Bx2[4][4];   // W_ih2: K=16 (h1), gate tiles padded 10 -> 16 cols
    v2f Bh2[4][4];   // W_hh2: K padded 10 -> 16, cols padded 10 -> 16

    #pragma unroll
    for (int kc = 0; kc < 8; ++kc)
        #pragma unroll
        for (int nt = 0; nt < 4; ++nt) {
            v2f v = *(const v2f*)(w_ih1 + (nt * 16 + col) * IN_SZ + 4 * kc + 2 * kh);
            float s = (nt == 2) ? 1.0f : 0.5f;
            Bx1[kc][nt] = (v2f){v.x * s, v.y * s};
        }

    #pragma unroll
    for (int kc = 0; kc < 4; ++kc)
        #pragma unroll
        for (int nt = 0; nt < 4; ++nt) {
            v2f v = *(const v2f*)(w_hh1 + (nt * 16 + col) * H1_SZ + 4 * kc + 2 * kh);
            float s = (nt == 2) ? 1.0f : 0.5f;
            Bh1[kc][nt] = (v2f){v.x * s, v.y * s};
        }

    #pragma unroll
    for (int kc = 0; kc < 4; ++kc)
        #pragma unroll
        for (int nt = 0; nt < 4; ++nt) {
            v2f v = (v2f){0.0f, 0.0f};
            if (col < H2_SZ)
                v = *(const v2f*)(w_ih2 + (nt * H2_SZ + col) * H1_SZ + 4 * kc + 2 * kh);
            float s = (nt == 2) ? 1.0f : 0.5f;
            Bx2[kc][nt] = (v2f){v.x * s, v.y * s};
        }

    #pragma unroll
    for (int kc = 0; kc < 4; ++kc)
        #pragma unroll
        for (int nt = 0; nt < 4; ++nt) {
            float v0 = 0.0f, v1 = 0.0f;
            int k0 = 4 * kc + 2 * kh;
            if (col < H2_SZ) {
                const float* wr = w_hh2 + (nt * H2_SZ + col) * H2_SZ;
                if (k0 < H2_SZ)     v0 = wr[k0];
                if (k0 + 1 < H2_SZ) v1 = wr[k0 + 1];
            }
            float s = (nt == 2) ? 1.0f : 0.5f;
            Bh2[kc][nt] = (v2f){v0 * s, v1 * s};
        }

    // Biases (per lane: gate tile nt, hidden unit col), same 0.5 pre-scale on i/f/o
    float bias1[4], bias2[4];
    #pragma unroll
    for (int nt = 0; nt < 4; ++nt) {
        float s = (nt == 2) ? 1.0f : 0.5f;
        bias1[nt] = (b_ih1[nt * 16 + col] + b_hh1[nt * 16 + col]) * s;
        bias2[nt] = (col < H2_SZ)
                  ? (b_ih2[nt * H2_SZ + col] + b_hh2[nt * H2_SZ + col]) * s
                  : 0.0f;
    }

    // ---- Recurrent state
    v2f h1A[4], h2A[4];              // h in A-fragment layout (lane = batch row col, K across frags)
    float c1r[8], c2r[8];            // cell state in C/D layout (rows r+8*kh, hidden col)
    #pragma unroll
    for (int kc = 0; kc < 4; ++kc) { h1A[kc] = (v2f){0.f, 0.f}; h2A[kc] = (v2f){0.f, 0.f}; }
    #pragma unroll
    for (int r = 0; r < 8; ++r) { c1r[r] = 0.0f; c2r[r] = 0.0f; }

    const float* xbase = x + (size_t)(blockIdx.x * 16 + col) * (size_t)(T_LEN * IN_SZ) + 2 * kh;

    for (int t = 0; t < T_LEN; ++t) {
        const float* xr = xbase + t * IN_SZ;
        if (t + 1 < T_LEN)
            __builtin_prefetch(xr + IN_SZ, 0, 3);   // global_prefetch of next timestep's row

        // ---- Layer 1 gates: (16xIN) x W_ih1^T  +  (16xH1) x W_hh1^T  + bias
        v8f g1[4];
        #pragma unroll
        for (int nt = 0; nt < 4; ++nt) {
            float b = bias1[nt];
            g1[nt] = (v8f){b, b, b, b, b, b, b, b};
        }
        #pragma unroll
        for (int kc = 0; kc < 8; ++kc) {
            v2f a = *(const v2f*)(xr + 4 * kc);     // x A-fragment, K chunk kc
            #pragma unroll
            for (int nt = 0; nt < 4; ++nt)
                g1[nt] = __builtin_amdgcn_wmma_f32_16x16x4_f32(
                    false, a, false, Bx1[kc][nt], (short)0, g1[nt], false, false);
        }
        #pragma unroll
        for (int kc = 0; kc < 4; ++kc)
            #pragma unroll
            for (int nt = 0; nt < 4; ++nt)
                g1[nt] = __builtin_amdgcn_wmma_f32_16x16x4_f32(
                    false, h1A[kc], false, Bh1[kc][nt], (short)0, g1[nt], false, false);

        // ---- Layer 1 cell (lane-local; C/D layout = lane holds hidden col, rows r+8*kh)
        #pragma unroll
        for (int r = 0; r < 8; ++r) {
            float ig = sigmoid_from_half(g1[0][r]);
            float fg = sigmoid_from_half(g1[1][r]);
            float gg = fast_tanh(g1[2][r]);
            float og = sigmoid_from_half(g1[3][r]);
            float c  = fg * c1r[r] + ig * gg;
            c1r[r]   = c;
            lds1[r + 8 * kh][col] = og * fast_tanh(c);   // transpose staging
        }
        // h1 -> A-fragment layout (lane = batch row `col`, K = 4*kc + 2*kh .. +1)
        #pragma unroll
        for (int kc = 0; kc < 4; ++kc)
            h1A[kc] = *(const v2f*)&lds1[col][4 * kc + 2 * kh];

        // ---- Layer 2 gates: h1 x W_ih2^T + h2 x W_hh2^T + bias (padded to 16x64)
        v8f g2[4];
        #pragma unroll
        for (int nt = 0; nt < 4; ++nt) {
            float b = bias2[nt];
            g2[nt] = (v8f){b, b, b, b, b, b, b, b};
        }
        #pragma unroll
        for (int kc = 0; kc < 4; ++kc)
            #pragma unroll
            for (int nt = 0; nt < 4; ++nt)
                g2[nt] = __builtin_amdgcn_wmma_f32_16x16x4_f32(
                    false, h1A[kc], false, Bx2[kc][nt], (short)0, g2[nt], false, false);
        #pragma unroll
        for (int kc = 0; kc < 4; ++kc)
            #pragma unroll
            for (int nt = 0; nt < 4; ++nt)
                g2[nt] = __builtin_amdgcn_wmma_f32_16x16x4_f32(
                    false, h2A[kc], false, Bh2[kc][nt], (short)0, g2[nt], false, false);

        // ---- Layer 2 cell; keep padded hidden units (col >= 10) at zero
        #pragma unroll
        for (int r = 0; r < 8; ++r) {
            float ig = sigmoid_from_half(g2[0][r]);
            float fg = sigmoid_from_half(g2[1][r]);
            float gg = fast_tanh(g2[2][r]);
            float og = sigmoid_from_half(g2[3][r]);
            float c  = fg * c2r[r] + ig * gg;
            c2r[r]   = c;
            float hv = og * fast_tanh(c);
            lds2[r + 8 * kh][col] = (col < H2_SZ) ? hv : 0.0f;
        }
        #pragma unroll
        for (int kc = 0; kc < 4; ++kc)
            h2A[kc] = *(const v2f*)&lds2[col][4 * kc + 2 * kh];
    }

    // ---- FC head on final h2 (lds2 holds h2[T-1]); lanes 0..15 -> one batch row each
    if (lane < 16) {
        float hv[H2_SZ];
        #pragma unroll
        for (int j = 0; j < H2_SZ; ++j) hv[j] = lds2[lane][j];

        float acc2 = fc2_b[0];
        #pragma unroll
        for (int k = 0; k < 5; ++k) {
            float a = fc1_b[k];
            #pragma unroll
            for (int j = 0; j < H2_SZ; ++j) a += fc1_w[k * H2_SZ + j] * hv[j];
            a = fmaxf(a, 0.0f);
            acc2 += fc2_w[k] * a;
        }
        // plain sigmoid on an unscaled logit
        acc2 = 0.5f * acc2;
        out[blockIdx.x * 16 + lane] = sigmoid_from_half(acc2);
    }
}

extern "C" void kernel_launch(void* const* d_in, const int* in_sizes, int n_in,
                              void* d_out, int out_size, void* d_ws, size_t ws_size,
                              hipStream_t stream) {
    (void)in_sizes; (void)n_in; (void)d_ws; (void)ws_size; (void)out_size;
    const float* x     = (const float*)d_in[0];
    const float* w_ih1 = (const float*)d_in[1];
    const float* w_hh1 = (const float*)d_in[2];
    const float* b_ih1 = (const float*)d_in[3];
    const float* b_hh1 = (const float*)d_in[4];
    const float* w_ih2 = (const float*)d_in[5];
    const float* w_hh2 = (const float*)d_in[6];
    const float* b_ih2 = (const float*)d_in[7];
    const float* b_hh2 = (const float*)d_in[8];
    const float* fc1_w = (const float*)d_in[9];
    const float* fc1_b = (const float*)d_in[10];
    const float* fc2_w = (const float*)d_in[11];
    const float* fc2_b = (const float*)d_in[12];
    float* out = (float*)d_out;

    dim3 grid(B_TOT / 16);
    dim3 block(32);
    lstm2_wmma_kernel<<<grid, block, 0, stream>>>(
        x, w_ih1, w_hh1, b_ih1, b_hh1,
        w_ih2, w_hh2, b_ih2, b_hh2,
        fc1_w, fc1_b, fc2_w, fc2_b, out);
}